// SelectiveSSM_51281909514844
// MI455X (gfx1250) — compile-verified
//
#include <hip/hip_runtime.h>
#include <stddef.h>
#include <stdint.h>

#define D_MODEL 1024
#define D_STATE 16
#define D_CONV  4
#define D_INNER 2048
#define DT_RANK 64
#define BATCH   2
#define SEQLEN  4096
#define ROWS    (BATCH * SEQLEN)
#define XP_COLS 96      // dt_rank + 2*d_state
#define XP_LD   128     // padded leading dimension (guard-free GEMM tiles)

typedef float v2f __attribute__((ext_vector_type(2)));
typedef float v8f __attribute__((ext_vector_type(8)));

__device__ __forceinline__ float sigmoidf_(float x) {
    return 1.0f / (1.0f + __expf(-x));
}

// Async global -> LDS copy of 16 bytes per lane (tracked by ASYNCcnt).
// LDS dest VGPR holds the wave-relative LDS byte offset (generic ptr low 32b).
__device__ __forceinline__ void async_b128(uint32_t lds_off, const float* gaddr) {
    asm volatile("global_load_async_to_lds_b128 %0, %1, off"
                 :: "v"(lds_off), "v"(gaddr) : "memory");
}
__device__ __forceinline__ void wait_async0() {
    asm volatile("s_wait_asynccnt 0x0" ::: "memory");
}

// ---------------------------------------------------------------------------
// Guard-free fp32 WMMA GEMM: C[M,N] = A[M,K] @ B[K,N] (row-major).
// Requires M%128==0, N%128==0, K%16==0 (arranged by buffer padding).
// Block tile 128x128, 256 threads = 8 waves (4 in M x 2 in N); each wave
// computes 32x64 as 2x4 V_WMMA_F32_16X16X4_F32 tiles. Double-buffered LDS
// fed by global_load_async_to_lds_b128 (ASYNCcnt pipeline).
// MODE 0: none.  MODE 1: softplus(acc + bias[col]).
// ---------------------------------------------------------------------------
template <int MODE>
__global__ __launch_bounds__(256) void gemm_wmma_f32(
    const float* __restrict__ A, const float* __restrict__ Bm,
    float* __restrict__ C, const float* __restrict__ bias,
    int M, int N, int K, int lda, int ldb, int ldc)
{
    // Row stride 20 floats = 80B: keeps B128 LDS writes 16B-aligned and the
    // 16-lane stride-20 fragment reads conflict-free (gcd(20,64)=4, 16 lanes
    // hit 16 distinct banks).
    __shared__ float lA[2][128][20];
    __shared__ float lB[2][16][128];

    const int tid  = threadIdx.x;
    const int lane = tid & 31;
    const int wave = tid >> 5;
    const int half = lane >> 4;      // 0: lanes 0-15, 1: lanes 16-31
    const int l16  = lane & 15;
    const int m0 = blockIdx.y * 128;
    const int n0 = blockIdx.x * 128;
    const int waveM = (wave >> 1) * 32;   // 4 waves in M
    const int waveN = (wave & 1) * 64;    // 2 waves in N

    v8f acc[2][4] = {};
    const int nk = K >> 4;

    auto prefetch = [&](int kt, int buf) {
        const int k0 = kt << 4;
        // A tile 128x16 floats: 512 B128 chunks, 2 per thread.
        #pragma unroll
        for (int i = 0; i < 2; ++i) {
            int c = tid + i * 256;
            int r = c >> 2, co = (c & 3) * 4;
            async_b128((uint32_t)(uintptr_t)&lA[buf][r][co],
                       A + (size_t)(m0 + r) * lda + (k0 + co));
        }
        // B tile 16x128 floats: 512 B128 chunks, 2 per thread.
        #pragma unroll
        for (int i = 0; i < 2; ++i) {
            int c = tid + i * 256;
            int r = c >> 5, co = (c & 31) * 4;
            async_b128((uint32_t)(uintptr_t)&lB[buf][r][co],
                       Bm + (size_t)(k0 + r) * ldb + (n0 + co));
        }
    };

    prefetch(0, 0);
    for (int kt = 0; kt < nk; ++kt) {
        const int cur = kt & 1;
        wait_async0();        // this wave's async writes to LDS complete
        __syncthreads();      // -> all waves' writes complete, tile kt ready
        if (kt + 1 < nk) prefetch(kt + 1, cur ^ 1);  // stream next tile

        #pragma unroll
        for (int kk = 0; kk < 4; ++kk) {
            // ISA 7.12.2 fp32 fragments: A 16x4: lanes 0-15 K=0,1; lanes
            // 16-31 K=2,3. B 4x16 row-striped symmetrically across halves.
            const int cb = kk * 4 + half * 2;
            v2f af[2], bf[4];
            #pragma unroll
            for (int tm = 0; tm < 2; ++tm) {
                int ar = waveM + tm * 16 + l16;
                af[tm][0] = lA[cur][ar][cb];
                af[tm][1] = lA[cur][ar][cb + 1];
            }
            #pragma unroll
            for (int tn = 0; tn < 4; ++tn) {
                int bc = waveN + tn * 16 + l16;
                bf[tn][0] = lB[cur][cb][bc];
                bf[tn][1] = lB[cur][cb + 1][bc];
            }
            #pragma unroll
            for (int tm = 0; tm < 2; ++tm)
                #pragma unroll
                for (int tn = 0; tn < 4; ++tn)
                    acc[tm][tn] = __builtin_amdgcn_wmma_f32_16x16x4_f32(
                        false, af[tm], false, bf[tn],
                        (short)0, acc[tm][tn], false, false);
        }
        __syncthreads();      // all reads of buf `cur` done before reuse
    }

    // C/D layout: VGPR r -> row (half*8 + r), col = l16 within a 16x16 tile.
    #pragma unroll
    for (int tm = 0; tm < 2; ++tm) {
        #pragma unroll
        for (int tn = 0; tn < 4; ++tn) {
            const int col = n0 + waveN + tn * 16 + l16;
            const float bv = (MODE == 1) ? bias[col] : 0.0f;
            #pragma unroll
            for (int r = 0; r < 8; ++r) {
                const int row = m0 + waveM + tm * 16 + half * 8 + r;
                float v = acc[tm][tn][r];
                if (MODE == 1) {
                    v += bv;
                    v = (v > 20.0f) ? v : log1pf(__expf(v));
                }
                C[(size_t)row * ldc + col] = v;
            }
        }
    }
}

// ---------------------------------------------------------------------------
// Zero-pad W_x (2048x96) into Wxp (2048x128) so the xp GEMM is guard-free.
// ---------------------------------------------------------------------------
__global__ __launch_bounds__(256) void pad_wx_kernel(
    const float* __restrict__ W_x, float* __restrict__ Wxp)
{
    int idx = blockIdx.x * blockDim.x + threadIdx.x;   // D_INNER * XP_LD
    int c = idx & (XP_LD - 1), k = idx >> 7;
    Wxp[idx] = (c < XP_COLS) ? W_x[k * XP_COLS + c] : 0.0f;
}

// ---------------------------------------------------------------------------
// Depthwise causal conv1d (width 4) + bias + SiLU, reading xi = xz[:, :2048].
// ---------------------------------------------------------------------------
__global__ __launch_bounds__(256) void conv_silu_kernel(
    const float* __restrict__ xzb, const float* __restrict__ w,
    const float* __restrict__ b, float* __restrict__ xcb)
{
    size_t idx = (size_t)blockIdx.x * blockDim.x + threadIdx.x;
    int d = (int)(idx % D_INNER);
    size_t row = idx / D_INNER;           // row = bb*SEQLEN + t
    int t = (int)(row % SEQLEN);
    float acc = b[d];
    #pragma unroll
    for (int k = 0; k < D_CONV; ++k) {
        int tt = t - (D_CONV - 1) + k;
        if (tt >= 0) {
            long rr = (long)row - (D_CONV - 1) + k;
            acc = __fmaf_rn(xzb[(size_t)rr * (2 * D_INNER) + d],
                            w[d * D_CONV + k], acc);
        }
    }
    xcb[idx] = acc * sigmoidf_(acc);
}

// ---------------------------------------------------------------------------
// Selective scan fused with skip (D) and SiLU(z) gate.
// One wave = 2 channels; 16 lanes per channel hold the 16 states.
// h_t = exp(dt*A)*h_{t-1} + dt*B_t*x_t ; y = <h,C> via shfl-xor butterfly.
// (stop_gradient between chunks is a no-op in the forward pass.)
// ---------------------------------------------------------------------------
__global__ __launch_bounds__(256) void scan_kernel(
    const float* __restrict__ dtb, const float* __restrict__ xcb,
    const float* __restrict__ xpb, const float* __restrict__ xzb,
    const float* __restrict__ A_log, const float* __restrict__ Dvec,
    float* __restrict__ yb)
{
    const int gtid   = blockIdx.x * blockDim.x + threadIdx.x;
    const int waveId = gtid >> 5;
    const int lane   = threadIdx.x & 31;
    const int half   = lane >> 4;
    const int s      = lane & 15;

    const int pairs = D_INNER / 2;               // 1024
    const int bb = waveId / pairs;
    const int pr = waveId % pairs;
    const int d  = pr * 2 + half;

    const float Aneg = -__expf(A_log[d * D_STATE + s]);   // A = -exp(A_log)
    const float Dd   = Dvec[d];
    float h = 0.0f;

    const size_t base = (size_t)bb * SEQLEN;
    for (int t = 0; t < SEQLEN; ++t) {
        const size_t row = base + t;
        const float dtv = dtb[row * D_INNER + d];    // broadcast in half-wave
        const float xv  = xcb[row * D_INNER + d];
        const float bv  = xpb[row * XP_LD + DT_RANK + s];
        const float cv  = xpb[row * XP_LD + DT_RANK + D_STATE + s];

        const float dA = __expf(dtv * Aneg);
        h = __fmaf_rn(dA, h, dtv * bv * xv);
        float p = h * cv;
        p += __shfl_xor(p, 1);
        p += __shfl_xor(p, 2);
        p += __shfl_xor(p, 4);
        p += __shfl_xor(p, 8);                       // sum over 16 states
        if (s == 0) {
            const float zv = xzb[row * (2 * D_INNER) + D_INNER + d];
            const float yv = (p + xv * Dd) * (zv * sigmoidf_(zv));
            yb[row * D_INNER + d] = yv;
        }
    }
}

// ---------------------------------------------------------------------------
extern "C" void kernel_launch(void* const* d_in, const int* in_sizes, int n_in,
                              void* d_out, int out_size, void* d_ws, size_t ws_size,
                              hipStream_t stream) {
    const float* x      = (const float*)d_in[0];
    const float* W_in   = (const float*)d_in[1];
    const float* conv_w = (const float*)d_in[2];
    const float* conv_b = (const float*)d_in[3];
    const float* W_x    = (const float*)d_in[4];
    const float* W_dt   = (const float*)d_in[5];
    const float* b_dt   = (const float*)d_in[6];
    const float* A_log  = (const float*)d_in[7];
    const float* Dvec   = (const float*)d_in[8];
    const float* W_out  = (const float*)d_in[9];
    float* out = (float*)d_out;

    char* wsb = (char*)d_ws;
    size_t off = 0;
    auto wsalloc = [&](size_t bytes) -> void* {
        void* p = wsb + off;
        off = (off + bytes + 255) & ~(size_t)255;
        return p;
    };
    float* xz  = (float*)wsalloc((size_t)ROWS * 2 * D_INNER * sizeof(float)); // 128 MB
    float* xc  = (float*)wsalloc((size_t)ROWS * D_INNER * sizeof(float));     // 64 MB
    float* xp  = (float*)wsalloc((size_t)ROWS * XP_LD * sizeof(float));       // 4 MB
    float* dtb = (float*)wsalloc((size_t)ROWS * D_INNER * sizeof(float));     // 64 MB
    float* yb  = (float*)wsalloc((size_t)ROWS * D_INNER * sizeof(float));     // 64 MB
    float* Wxp = (float*)wsalloc((size_t)D_INNER * XP_LD * sizeof(float));    // 1 MB
    (void)ws_size; (void)in_sizes; (void)n_in; (void)out_size;

    // 0) pad W_x (2048x96 -> 2048x128, zero-filled)
    pad_wx_kernel<<<(D_INNER * XP_LD) / 256, 256, 0, stream>>>(W_x, Wxp);

    // 1) xz = x @ W_in   (8192x1024 @ 1024x4096)
    {
        dim3 g((2 * D_INNER) / 128, ROWS / 128);
        gemm_wmma_f32<0><<<g, 256, 0, stream>>>(x, W_in, xz, nullptr,
            ROWS, 2 * D_INNER, D_MODEL, D_MODEL, 2 * D_INNER, 2 * D_INNER);
    }
    // 2) xc = silu(causal_conv(xi) + conv_b)
    {
        size_t total = (size_t)ROWS * D_INNER;
        conv_silu_kernel<<<(int)(total / 256), 256, 0, stream>>>(xz, conv_w, conv_b, xc);
    }
    // 3) xp = xc @ Wxp   (8192x2048 @ 2048x128, cols 96..127 are zero)
    {
        dim3 g(XP_LD / 128, ROWS / 128);
        gemm_wmma_f32<0><<<g, 256, 0, stream>>>(xc, Wxp, xp, nullptr,
            ROWS, XP_LD, D_INNER, D_INNER, XP_LD, XP_LD);
    }
    // 4) dt = softplus(xp[:, :64] @ W_dt + b_dt)   (K=64, lda=128)
    {
        dim3 g(D_INNER / 128, ROWS / 128);
        gemm_wmma_f32<1><<<g, 256, 0, stream>>>(xp, W_dt, dtb, b_dt,
            ROWS, D_INNER, DT_RANK, XP_LD, D_INNER, D_INNER);
    }
    // 5) selective scan + skip + gate -> yb
    {
        int waves = BATCH * (D_INNER / 2);           // 2048 waves
        int threads = waves * 32;                    // 65536
        scan_kernel<<<threads / 256, 256, 0, stream>>>(dtb, xc, xp, xz,
                                                       A_log, Dvec, yb);
    }
    // 6) out = yb @ W_out   (8192x2048 @ 2048x1024)
    {
        dim3 g(D_MODEL / 128, ROWS / 128);
        gemm_wmma_f32<0><<<g, 256, 0, stream>>>(yb, W_out, out, nullptr,
            ROWS, D_MODEL, D_INNER, D_INNER, D_MODEL, D_MODEL);
    }
}